// CGCN_84043920048189
// MI455X (gfx1250) — compile-verified
//
#include <hip/hip_runtime.h>
#include <math.h>

// Problem constants (from reference)
#define B_  4
#define N_  512
#define F_  128
#define H_  4
#define L_  2
#define HID_ 128
#define LAM_ 5.0f
#define ITERS_ 10
#define NEG_SLOPE_ 0.2f

typedef __attribute__((ext_vector_type(2))) float v2f;
typedef __attribute__((ext_vector_type(8))) float v8f;

// ---------------------------------------------------------------------------
// One wave computes a 16x16 f32 tile of C = A(16xK) * B(Kx16) using
// V_WMMA_F32_16X16X4_F32, looping K in steps of 4.
// A is row-major with leading dim lda.
// BT=false: B is KxN row-major (ldb = row stride). BT=true: B is NxK row-major
// (i.e. we compute A * B^T), ldb = row stride of B.
// Per ISA layout (32-bit A 16x4): lanes 0-15 hold K={0,1} in v0/v1, lanes
// 16-31 hold K={2,3}. B mirrors with N=lane index. C/D: VGPR r holds row r
// (lanes 0-15) and row r+8 (lanes 16-31).
// ---------------------------------------------------------------------------
template <bool BT>
__device__ inline v8f wmma_tile_f32(const float* __restrict__ A, int lda,
                                    const float* __restrict__ Bm, int ldb,
                                    int K, v8f acc) {
  const int lane = threadIdx.x & 31;
  const int half = lane >> 4;
  const int l16  = lane & 15;
  const int koff = 2 * half;
  #pragma unroll 4
  for (int k = 0; k < K; k += 4) {
    v2f a, b;
    a[0] = A[l16 * lda + k + koff];
    a[1] = A[l16 * lda + k + koff + 1];
    if (BT) {
      b[0] = Bm[l16 * ldb + k + koff];
      b[1] = Bm[l16 * ldb + k + koff + 1];
    } else {
      b[0] = Bm[(k + koff) * ldb + l16];
      b[1] = Bm[(k + koff + 1) * ldb + l16];
    }
    acc = __builtin_amdgcn_wmma_f32_16x16x4_f32(
        /*neg_a=*/false, a, /*neg_b=*/false, b,
        /*c_mod=*/(short)0, acc, /*reuse_a=*/false, /*reuse_b=*/false);
  }
  return acc;
}

__device__ inline void wmma_store_tile(float* __restrict__ C, int ldc, v8f acc) {
  const int lane = threadIdx.x & 31;
  const int half = lane >> 4;
  const int l16  = lane & 15;
  #pragma unroll
  for (int r = 0; r < 8; ++r)
    C[(r + 8 * half) * ldc + l16] = acc[r];
}

// ---------------------------------------------------------------------------
// K1: ha = features @ W1[:F],  hb = features @ W1[F:2F]
// grid: (B*N/16, HID/16, 2), block 32 (one wave per 16x16 tile)
// ---------------------------------------------------------------------------
__global__ __launch_bounds__(32)
void k_ha_hb(const float* __restrict__ jf, const float* __restrict__ W1,
             float* __restrict__ ha, float* __restrict__ hb) {
  const int mt = blockIdx.x, nt = blockIdx.y, which = blockIdx.z;
  const float* A  = jf + (size_t)mt * 16 * F_;
  const float* Bm = W1 + (size_t)which * F_ * HID_ + nt * 16;
  v8f acc = {};
  acc = wmma_tile_f32<false>(A, F_, Bm, HID_, F_, acc);
  float* C = (which == 0 ? ha : hb) + (size_t)mt * 16 * HID_ + nt * 16;
  wmma_store_tile(C, HID_, acc);
}

// ---------------------------------------------------------------------------
// K2: cost[b,i,j] = sum_k relu(ha+hb+dist*wd+b1)*W2 ; P0 = exp(-LAM*(cost+b2))
// ReLU inside the reduction -> pure VALU; tile (16x16) of (i,j) per block,
// ha/hb rows + wd/b1/W2 staged in LDS.
// grid: (N/16, N/16, B), block 256
// ---------------------------------------------------------------------------
__global__ __launch_bounds__(256)
void k_cost(const float* __restrict__ ha, const float* __restrict__ hb,
            const float* __restrict__ coords, const float* __restrict__ W1,
            const float* __restrict__ b1, const float* __restrict__ W2,
            const float* __restrict__ b2, float* __restrict__ adj) {
  __shared__ float sha[16][HID_ + 1];
  __shared__ float shb[16][HID_ + 1];
  __shared__ float swd[HID_], sb1[HID_], sw2[HID_];
  const int b  = blockIdx.z;
  const int i0 = blockIdx.x * 16, j0 = blockIdx.y * 16;
  const int tid = threadIdx.x;
  for (int idx = tid; idx < 16 * HID_; idx += 256) {
    int r = idx >> 7, c = idx & (HID_ - 1);
    sha[r][c] = ha[((size_t)(b * N_ + i0 + r)) * HID_ + c];
    shb[r][c] = hb[((size_t)(b * N_ + j0 + r)) * HID_ + c];
  }
  for (int idx = tid; idx < HID_; idx += 256) {
    swd[idx] = W1[2 * F_ * HID_ + idx];  // W1 row 2F = wd
    sb1[idx] = b1[idx];
    sw2[idx] = W2[idx];
  }
  __syncthreads();
  const int ti = tid >> 4, tj = tid & 15;
  const int i = i0 + ti, j = j0 + tj;
  const float dx = coords[(size_t)(b * N_ + i) * 3 + 0] - coords[(size_t)(b * N_ + j) * 3 + 0];
  const float dy = coords[(size_t)(b * N_ + i) * 3 + 1] - coords[(size_t)(b * N_ + j) * 3 + 1];
  const float dz = coords[(size_t)(b * N_ + i) * 3 + 2] - coords[(size_t)(b * N_ + j) * 3 + 2];
  const float dist = sqrtf(fmaxf(dx * dx + dy * dy + dz * dz, 0.0f));
  float cost = 0.0f;
  #pragma unroll 4
  for (int k = 0; k < HID_; ++k) {
    float t = sha[ti][k] + shb[tj][k] + dist * swd[k] + sb1[k];
    cost = fmaf(fmaxf(t, 0.0f), sw2[k], cost);
  }
  cost += b2[0];
  adj[((size_t)b * N_ + i) * N_ + j] = __expf(-LAM_ * cost);
}

// ---------------------------------------------------------------------------
// K3: fused Sinkhorn. P after k steps equals diag(r) * P0 * diag(c) with
//   r <- 1/(P0 c),  c <- 1/(P0^T r)  per step, c initialized to ones.
// One 512-thread workgroup per batch; r/c live in LDS; workgroup barriers
// between matvec phases. P0 (1 MB/batch) stays L2-resident.
// ---------------------------------------------------------------------------
__global__ __launch_bounds__(512)
void k_sinkvec(const float* __restrict__ P0, float* __restrict__ rvec,
               float* __restrict__ cvec) {
  __shared__ float sr[N_];
  __shared__ float sc[N_];
  const int b = blockIdx.x;
  const int t = threadIdx.x;
  const float* P = P0 + (size_t)b * N_ * N_;
  sc[t] = 1.0f;
  __syncthreads();
  for (int it = 0; it < ITERS_; ++it) {
    // r = 1 / (P0 c): thread t owns row t (streams one 2KB row, L2-resident)
    const float* row = P + (size_t)t * N_;
    float s = 0.0f;
    #pragma unroll 4
    for (int j = 0; j < N_; ++j) s = fmaf(row[j], sc[j], s);
    sr[t] = 1.0f / s;
    __syncthreads();
    // c = 1 / (P0^T r): thread t owns column t (coalesced across threads)
    float s2 = 0.0f;
    #pragma unroll 4
    for (int i = 0; i < N_; ++i) s2 = fmaf(P[(size_t)i * N_ + t], sr[i], s2);
    sc[t] = 1.0f / s2;
    __syncthreads();
  }
  rvec[b * N_ + t] = sr[t];
  cvec[b * N_ + t] = sc[t];
}

// K3b: apply rank-1 scaling in place: P[b,i,j] *= r[b,i]*c[b,j]
// grid: (N, B), block 512
__global__ __launch_bounds__(512)
void k_applyrc(float* __restrict__ P, const float* __restrict__ rvec,
               const float* __restrict__ cvec) {
  const int i = blockIdx.x, b = blockIdx.y;
  float* p = P + ((size_t)b * N_ + i) * N_;
  const float r = rvec[b * N_ + i];
  const int j = threadIdx.x;
  p[j] *= r * cvec[b * N_ + j];
}

// ---------------------------------------------------------------------------
// K5: xw = x @ Wg[l]   (B*N x F) @ (F x H*F)
// ---------------------------------------------------------------------------
__global__ __launch_bounds__(32)
void k_gemm_xw(const float* __restrict__ x, const float* __restrict__ Wg_l,
               float* __restrict__ xw) {
  const int mt = blockIdx.x, nt = blockIdx.y;
  v8f acc = {};
  acc = wmma_tile_f32<false>(x + (size_t)mt * 16 * F_, F_,
                             Wg_l + nt * 16, H_ * F_, F_, acc);
  wmma_store_tile(xw + (size_t)mt * 16 * (H_ * F_) + nt * 16, H_ * F_, acc);
}

// ---------------------------------------------------------------------------
// K5b: s_src[b,n,h] = xw . a_src[h],  s_dst likewise
// ---------------------------------------------------------------------------
__global__ __launch_bounds__(256)
void k_scores(const float* __restrict__ xw, const float* __restrict__ a_src_l,
              const float* __restrict__ a_dst_l,
              float* __restrict__ s_src, float* __restrict__ s_dst) {
  const int idx = blockIdx.x * blockDim.x + threadIdx.x;
  if (idx >= B_ * N_ * H_) return;
  const int h  = idx & (H_ - 1);
  const int bn = idx >> 2;
  const float* row = xw + (size_t)bn * H_ * F_ + h * F_;
  const float* as  = a_src_l + h * F_;
  const float* ad  = a_dst_l + h * F_;
  float ss = 0.0f, sd = 0.0f;
  #pragma unroll 4
  for (int f = 0; f < F_; ++f) {
    const float v = row[f];
    ss = fmaf(v, as[f], ss);
    sd = fmaf(v, ad[f], sd);
  }
  s_src[idx] = ss;
  s_dst[idx] = sd;
}

// ---------------------------------------------------------------------------
// K6: e = leaky_relu(s_dst[i] + s_src[j]); attn[b][h][i][:] = softmax_j(e)
// grid: (N, H, B), block 256 (2 elements/thread)
// ---------------------------------------------------------------------------
__global__ __launch_bounds__(256)
void k_attn(const float* __restrict__ s_src, const float* __restrict__ s_dst,
            float* __restrict__ attn) {
  __shared__ float red[256];
  const int i = blockIdx.x, h = blockIdx.y, b = blockIdx.z;
  const float sd = s_dst[(size_t)(b * N_ + i) * H_ + h];
  float e[2];
  float m = -1e30f;
  #pragma unroll
  for (int t = 0; t < 2; ++t) {
    const int j = threadIdx.x + t * 256;
    float v = sd + s_src[(size_t)(b * N_ + j) * H_ + h];
    v = v > 0.0f ? v : NEG_SLOPE_ * v;
    e[t] = v;
    m = fmaxf(m, v);
  }
  red[threadIdx.x] = m;
  __syncthreads();
  for (int st = 128; st > 0; st >>= 1) {
    if (threadIdx.x < st) red[threadIdx.x] = fmaxf(red[threadIdx.x], red[threadIdx.x + st]);
    __syncthreads();
  }
  m = red[0];
  __syncthreads();
  float s = 0.0f;
  #pragma unroll
  for (int t = 0; t < 2; ++t) { e[t] = __expf(e[t] - m); s += e[t]; }
  red[threadIdx.x] = s;
  __syncthreads();
  for (int st = 128; st > 0; st >>= 1) {
    if (threadIdx.x < st) red[threadIdx.x] += red[threadIdx.x + st];
    __syncthreads();
  }
  const float inv = 1.0f / red[0];
  float* out = attn + (((size_t)b * H_ + h) * N_ + i) * N_;
  #pragma unroll
  for (int t = 0; t < 2; ++t) {
    const int j = threadIdx.x + t * 256;
    out[j] = e[t] * inv;
  }
}

// ---------------------------------------------------------------------------
// K7: x_out[b,i,f] = (1/H) * sum_h (attn[b,h] @ xw[b,:,h,:])[i,f] + bg[l][f]
// grid: (N/16, F/16, B), block 32; K=512 WMMA loop per head, 4 heads summed
// ---------------------------------------------------------------------------
__global__ __launch_bounds__(32)
void k_aggregate(const float* __restrict__ attn, const float* __restrict__ xw,
                 const float* __restrict__ bg_l, float* __restrict__ xout) {
  const int it = blockIdx.x, ft = blockIdx.y, b = blockIdx.z;
  v8f acc = {};
  for (int h = 0; h < H_; ++h) {
    const float* A  = attn + (((size_t)b * H_ + h) * N_ + it * 16) * N_;
    const float* Bm = xw + (size_t)b * N_ * H_ * F_ + h * F_ + ft * 16;
    acc = wmma_tile_f32<false>(A, N_, Bm, H_ * F_, N_, acc);
  }
  const int lane = threadIdx.x & 31;
  const int half = lane >> 4;
  const int l16  = lane & 15;
  const float bgv = bg_l[ft * 16 + l16];
  float* C = xout + ((size_t)(b * N_) + it * 16) * F_ + ft * 16;
  #pragma unroll
  for (int r = 0; r < 8; ++r)
    C[(r + 8 * half) * F_ + l16] = acc[r] * (1.0f / H_) + bgv;
}

// ---------------------------------------------------------------------------
// K8: final_adj = sigmoid(x @ x^T) per batch (A * B^T WMMA variant)
// ---------------------------------------------------------------------------
__global__ __launch_bounds__(32)
void k_final(const float* __restrict__ x, float* __restrict__ out) {
  const int it = blockIdx.x, jt = blockIdx.y, b = blockIdx.z;
  const float* xb = x + (size_t)b * N_ * F_;
  v8f acc = {};
  acc = wmma_tile_f32<true>(xb + (size_t)it * 16 * F_, F_,
                            xb + (size_t)jt * 16 * F_, F_, F_, acc);
  const int lane = threadIdx.x & 31;
  const int half = lane >> 4;
  const int l16  = lane & 15;
  float* C = out + ((size_t)b * N_ + it * 16) * N_ + jt * 16;
  #pragma unroll
  for (int r = 0; r < 8; ++r)
    C[(r + 8 * half) * N_ + l16] = 1.0f / (1.0f + __expf(-acc[r]));
}

// ---------------------------------------------------------------------------
extern "C" void kernel_launch(void* const* d_in, const int* in_sizes, int n_in,
                              void* d_out, int out_size, void* d_ws, size_t ws_size,
                              hipStream_t stream) {
  const float* jf     = (const float*)d_in[0];
  const float* coords = (const float*)d_in[1];
  const float* W1     = (const float*)d_in[2];
  const float* b1     = (const float*)d_in[3];
  const float* W2     = (const float*)d_in[4];
  const float* b2     = (const float*)d_in[5];
  const float* Wg     = (const float*)d_in[6];
  const float* a_src  = (const float*)d_in[7];
  const float* a_dst  = (const float*)d_in[8];
  const float* bg     = (const float*)d_in[9];

  float* out       = (float*)d_out;
  float* final_adj = out;                          // B*N*N
  float* adj       = out + (size_t)B_ * N_ * N_;   // B*N*N (also holds P0)

  // workspace carve-up (floats)
  float* ws   = (float*)d_ws;
  float* ha   = ws;                                   // B*N*HID
  float* hb   = ha + (size_t)B_ * N_ * HID_;          // B*N*HID
  float* xw   = hb + (size_t)B_ * N_ * HID_;          // B*N*H*F
  float* ssrc = xw + (size_t)B_ * N_ * H_ * F_;       // B*N*H
  float* sdst = ssrc + (size_t)B_ * N_ * H_;          // B*N*H
  float* rvec = sdst + (size_t)B_ * N_ * H_;          // B*N
  float* cvec = rvec + (size_t)B_ * N_;               // B*N
  float* attb = cvec + (size_t)B_ * N_;               // B*H*N*N
  float* x1   = attb + (size_t)B_ * H_ * N_ * N_;     // B*N*F
  float* x2   = x1 + (size_t)B_ * N_ * F_;            // B*N*F

  // 1) ha / hb GEMMs (WMMA f32)
  k_ha_hb<<<dim3((B_ * N_) / 16, HID_ / 16, 2), 32, 0, stream>>>(jf, W1, ha, hb);

  // 2) cost + P0 = exp(-LAM*cost) (VALU, LDS-tiled), written to adj slot
  k_cost<<<dim3(N_ / 16, N_ / 16, B_), 256, 0, stream>>>(ha, hb, coords, W1, b1, W2, b2, adj);

  // 3) Sinkhorn fused: 10 iterations in one persistent kernel on r/c vectors,
  //    then one rank-1 apply in place on P0.
  k_sinkvec<<<B_, 512, 0, stream>>>(adj, rvec, cvec);
  k_applyrc<<<dim3(N_, B_), 512, 0, stream>>>(adj, rvec, cvec);

  // 4) GAT layers
  const float* xcur = jf;
  float* xnext = x1;
  for (int l = 0; l < L_; ++l) {
    k_gemm_xw<<<dim3((B_ * N_) / 16, (H_ * F_) / 16), 32, 0, stream>>>(
        xcur, Wg + (size_t)l * F_ * H_ * F_, xw);
    k_scores<<<(B_ * N_ * H_ + 255) / 256, 256, 0, stream>>>(
        xw, a_src + (size_t)l * H_ * F_, a_dst + (size_t)l * H_ * F_, ssrc, sdst);
    k_attn<<<dim3(N_, H_, B_), 256, 0, stream>>>(ssrc, sdst, attb);
    k_aggregate<<<dim3(N_ / 16, F_ / 16, B_), 32, 0, stream>>>(
        attb, xw, bg + (size_t)l * F_, xnext);
    xcur  = xnext;
    xnext = (xnext == x1) ? x2 : x1;
  }

  // 5) final_adj = sigmoid(x @ x^T)
  k_final<<<dim3(N_ / 16, N_ / 16, B_), 32, 0, stream>>>(xcur, final_adj);
}